// BaselineSSM_60155311948504
// MI455X (gfx1250) — compile-verified
//
#include <hip/hip_runtime.h>
#include <cstdint>

// ---------------------------------------------------------------------------
// SSM scan on MI455X (gfx1250, wave32).
//  - V_WMMA_F32_16X16X4_F32 for all matrix math (native f32 path; the 2048-step
//    recurrence is precision-sensitive, so no bf16 downcast).
//  - GLOBAL_LOAD_ASYNC_TO_LDS_B128 (+ s_wait_asynccnt) to stage shared act
//    tiles into LDS once per block instead of 8x per wave.
//  - Chunked parallel linear scan: 128 parallel local scans + 128-step carry
//    recurrence + power-of-A fixup (critical path 2048 -> ~160 steps).
// ---------------------------------------------------------------------------

typedef float v2f __attribute__((ext_vector_type(2)));
typedef float v8f __attribute__((ext_vector_type(8)));

constexpr int BATCH  = 16;
constexpr int SEQ    = 2048;
constexpr int HID    = 512;
constexpr int L      = 16;            // chunk length
constexpr int NCHUNK = SEQ / L;       // 128
constexpr int MROWS  = BATCH * SEQ;   // 32768 rows for the big GEMMs

// --- async global->LDS staging (CDNA5 path, tracked by ASYNCcnt) -----------
// Copies n16 * 16 bytes from g into LDS at lds_base, 256 threads cooperating.
__device__ __forceinline__ void async_stage(uint32_t lds_base,
                                            const float* __restrict__ g,
                                            int tid, int n16) {
  for (int j = tid; j < n16; j += 256) {
    const uint32_t off = (uint32_t)j * 16u;
    asm volatile("global_load_async_to_lds_b128 %0, %1, %2"
                 :: "v"(lds_base + off), "v"(off), "s"(g) : "memory");
  }
}

__device__ __forceinline__ void async_stage_wait() {
  asm volatile("s_wait_asynccnt 0x0" ::: "memory");
  __syncthreads();
}

// --- wave-level 16x16 tile helpers -----------------------------------------
// WMMA f32 16x16x4 operand layout (ISA 7.12.2):
//   A (16x4): lane l holds row l%16, K = 2*(l/16) + {0,1}  -> float2 per step
//   B (4x16): lane l holds col l%16, K = 2*(l/16) + {0,1}
//   C/D     : lane l holds col l%16, VGPR r -> row r + 8*(l/16)

// D += Act(16xK) @ W(rows=N, K)^T   (nn.Linear form, both row-major)
template<int NT>
__device__ __forceinline__ void wave_tile_wt(const float* act, int lda,
                                             const float* w,  int ldw,
                                             v8f* acc, int lane) {
  const int half = lane >> 4;
  const int sub  = lane & 15;
  const float* ap = act + sub * lda + 2 * half;
  const float* wp = w   + sub * ldw + 2 * half;
  #pragma unroll 8
  for (int k = 0; k < HID; k += 4) {
    v2f a = *(const v2f*)(ap + k);
    #pragma unroll
    for (int j = 0; j < NT; ++j) {
      v2f b = *(const v2f*)(wp + j * 16 * ldw + k);
      acc[j] = __builtin_amdgcn_wmma_f32_16x16x4_f32(
          false, a, false, b, (short)0, acc[j], false, false);
    }
  }
}

// D += Act(16xK) @ W(KxN)   (non-transposed W; used only for A-powers)
template<int NT>
__device__ __forceinline__ void wave_tile_nn(const float* act, int lda,
                                             const float* w,  int ldw,
                                             v8f* acc, int lane) {
  const int half = lane >> 4;
  const int sub  = lane & 15;
  const float* ap = act + sub * lda + 2 * half;
  #pragma unroll 4
  for (int k = 0; k < HID; k += 4) {
    v2f a = *(const v2f*)(ap + k);
    const int kr = k + 2 * half;
    #pragma unroll
    for (int j = 0; j < NT; ++j) {
      v2f b;
      b.x = w[(size_t)kr       * ldw + j * 16 + sub];
      b.y = w[(size_t)(kr + 1) * ldw + j * 16 + sub];
      acc[j] = __builtin_amdgcn_wmma_f32_16x16x4_f32(
          false, a, false, b, (short)0, acc[j], false, false);
    }
  }
}

__device__ __forceinline__ v8f load_tile(const float* in, int ldo, int lane) {
  const int half = lane >> 4, sub = lane & 15;
  v8f r;
  #pragma unroll
  for (int i = 0; i < 8; ++i) r[i] = in[(size_t)(i + 8 * half) * ldo + sub];
  return r;
}

__device__ __forceinline__ void store_tile(float* out, int ldo, v8f a, int lane) {
  const int half = lane >> 4, sub = lane & 15;
  #pragma unroll
  for (int i = 0; i < 8; ++i) out[(size_t)(i + 8 * half) * ldo + sub] = a[i];
}

// --- K1 / K6: out(Mx512) = act(Mx512) @ W(512x512)^T -----------------------
// grid = (M/32, 2), block = 256 (8 waves). 32x512 act tile async-staged into
// 64 KB LDS; waves = 2 M-tiles x 4 N-tile groups; each wave owns 16x64 output.
__global__ __launch_bounds__(256)
void gemm_wt_kernel(const float* __restrict__ act, const float* __restrict__ w,
                    float* __restrict__ out) {
  __shared__ float atile[32 * HID];                 // 64 KB
  const int tid  = threadIdx.x;
  const int wave = tid >> 5;
  const int lane = tid & 31;
  const int mt   = blockIdx.x;

  async_stage((uint32_t)(uintptr_t)atile, act + (size_t)mt * 32 * HID,
              tid, 32 * HID * 4 / 16);              // 4096 x 16B
  async_stage_wait();

  const int msub = wave >> 2;                        // 0..1 : M-tile in block
  const int n0   = (blockIdx.y * 4 + (wave & 3)) * 64;
  v8f acc[4] = {};
  wave_tile_wt<4>(atile + msub * 16 * HID, HID, w + (size_t)n0 * HID, HID,
                  acc, lane);
  #pragma unroll
  for (int j = 0; j < 4; ++j)
    store_tile(out + (size_t)(mt * 32 + msub * 16) * HID + n0 + j * 16, HID,
               acc[j], lane);
}

// --- K2: A-power chain: out = prev @ A  (512x512 each) ----------------------
__global__ __launch_bounds__(256)
void power_step_kernel(const float* __restrict__ prev, const float* __restrict__ A,
                       float* __restrict__ out) {
  const int wave = threadIdx.x >> 5;
  const int lane = threadIdx.x & 31;
  const int mt   = blockIdx.x;                   // 0..31
  const int n0   = (blockIdx.y * 8 + wave) * 32;
  v8f acc[2] = {};
  wave_tile_nn<2>(prev + (size_t)mt * 16 * HID, HID, A + n0, HID, acc, lane);
  #pragma unroll
  for (int j = 0; j < 2; ++j)
    store_tile(out + (size_t)mt * 16 * HID + n0 + j * 16, HID, acc[j], lane);
}

__global__ __launch_bounds__(256)
void copy_kernel(const float* __restrict__ src, float* __restrict__ dst, int n) {
  int i = blockIdx.x * blockDim.x + threadIdx.x;
  if (i < n) dst[i] = src[i];
}

// --- K3: per-chunk local scan (h0 = 0), in place over the u values in hs ----
// grid = NCHUNK, block = 256 (8 waves); h kept in 32 KB LDS; each wave owns
// 4 N-tiles (8 waves * 4 * 16 = 512 columns), M-dim = 16 batch rows.
__global__ __launch_bounds__(256)
void local_scan_kernel(float* __restrict__ hs, const float* __restrict__ A) {
  __shared__ float hbuf[BATCH * HID];           // 32 KB
  const int c    = blockIdx.x;
  const int wave = threadIdx.x >> 5;
  const int lane = threadIdx.x & 31;
  for (int i = threadIdx.x; i < BATCH * HID; i += blockDim.x) hbuf[i] = 0.f;
  __syncthreads();
  for (int t = 0; t < L; ++t) {
    const int s = c * L + t;
    v8f acc[4];
    #pragma unroll
    for (int j = 0; j < 4; ++j) {               // acc = u_t  (rows = batch)
      const int n0 = wave * 64 + j * 16;
      acc[j] = load_tile(hs + (size_t)s * HID + n0, SEQ * HID, lane);
    }
    // acc += h_prev @ A^T   (h_prev in LDS; zero at t==0)
    wave_tile_wt<4>(hbuf, HID, A + (size_t)(wave * 64) * HID, HID, acc, lane);
    __syncthreads();                            // all hbuf reads done
    #pragma unroll
    for (int j = 0; j < 4; ++j) {
      const int n0 = wave * 64 + j * 16;
      store_tile(hbuf + n0, HID, acc[j], lane);
      store_tile(hs + (size_t)s * HID + n0, SEQ * HID, acc[j], lane);
    }
    __syncthreads();                            // hbuf fully updated
  }
}

// --- K4: inter-chunk carry recurrence (only 128 sequential steps) ----------
// carry_c = local_end_c + carry_{c-1} @ (A^L)^T ; grid = 1, block = 256.
__global__ __launch_bounds__(256)
void carry_kernel(const float* __restrict__ hs, const float* __restrict__ Plast,
                  float* __restrict__ carr) {
  __shared__ float hbuf[BATCH * HID];
  const int wave = threadIdx.x >> 5;
  const int lane = threadIdx.x & 31;
  for (int i = threadIdx.x; i < BATCH * HID; i += blockDim.x) {
    const int b = i / HID, n = i % HID;
    const float v = hs[((size_t)b * SEQ + (L - 1)) * HID + n]; // carry_0
    hbuf[i] = v;
    carr[i] = v;
  }
  __syncthreads();
  for (int c = 1; c < NCHUNK; ++c) {
    const int s = c * L + (L - 1);
    v8f acc[4];
    #pragma unroll
    for (int j = 0; j < 4; ++j) {
      const int n0 = wave * 64 + j * 16;
      acc[j] = load_tile(hs + (size_t)s * HID + n0, SEQ * HID, lane);
    }
    wave_tile_wt<4>(hbuf, HID, Plast + (size_t)(wave * 64) * HID, HID, acc, lane);
    __syncthreads();
    #pragma unroll
    for (int j = 0; j < 4; ++j) {
      const int n0 = wave * 64 + j * 16;
      store_tile(hbuf + n0, HID, acc[j], lane);
      store_tile(carr + (size_t)c * BATCH * HID + n0, HID, acc[j], lane);
    }
    __syncthreads();
  }
}

// --- K5: fixup: hs[c*L+t] += carry_{c-1} @ (A^{t+1})^T  (fully parallel) ----
// grid = (NCHUNK-1, L), block = 256 (8 waves x 4 tiles = 512 columns).
// The 32 KB carry tile (shared by all waves) is async-staged into LDS.
__global__ __launch_bounds__(256)
void fixup_kernel(float* __restrict__ hs, const float* __restrict__ powers,
                  const float* __restrict__ carr) {
  __shared__ float ctile[BATCH * HID];          // 32 KB
  const int c    = blockIdx.x + 1;              // 1..127
  const int t    = blockIdx.y;                  // 0..15
  const int s    = c * L + t;
  const int tid  = threadIdx.x;
  const int wave = tid >> 5;
  const int lane = tid & 31;

  async_stage((uint32_t)(uintptr_t)ctile, carr + (size_t)(c - 1) * BATCH * HID,
              tid, BATCH * HID * 4 / 16);       // 2048 x 16B
  const float* P = powers + (size_t)t * HID * HID;
  v8f acc[4];
  #pragma unroll
  for (int j = 0; j < 4; ++j) {
    const int n0 = wave * 64 + j * 16;
    acc[j] = load_tile(hs + (size_t)s * HID + n0, SEQ * HID, lane);
  }
  async_stage_wait();
  wave_tile_wt<4>(ctile, HID, P + (size_t)(wave * 64) * HID, HID, acc, lane);
  #pragma unroll
  for (int j = 0; j < 4; ++j) {
    const int n0 = wave * 64 + j * 16;
    store_tile(hs + (size_t)s * HID + n0, SEQ * HID, acc[j], lane);
  }
}

// ---------------------------------------------------------------------------
extern "C" void kernel_launch(void* const* d_in, const int* in_sizes, int n_in,
                              void* d_out, int out_size, void* d_ws, size_t ws_size,
                              hipStream_t stream) {
  const float* x = (const float*)d_in[0];   // (16,2048,512)
  const float* A = (const float*)d_in[1];   // (512,512)
  const float* B = (const float*)d_in[2];   // (512,512)
  const float* C = (const float*)d_in[3];   // (512,512)

  float* ys = (float*)d_out;                       // (16,2048,512)
  float* hs = ys + (size_t)MROWS * HID;            // (16,2048,512)

  float* powers = (float*)d_ws;                    // L * 512*512 f32 = 16 MB
  float* carr   = powers + (size_t)L * HID * HID;  // NCHUNK * 16*512 f32 = 4 MB

  const dim3 blk(256);

  // K1: u = x @ B^T  -> hs region (in-place scan input)
  gemm_wt_kernel<<<dim3(MROWS / 32, 2), blk, 0, stream>>>(x, B, hs);

  // K2: powers P[t] = A^{t+1}  (P[0] = A, P[t] = P[t-1] @ A)
  copy_kernel<<<(HID * HID) / 256, blk, 0, stream>>>(A, powers, HID * HID);
  for (int t = 1; t < L; ++t)
    power_step_kernel<<<dim3(HID / 16, 2), blk, 0, stream>>>(
        powers + (size_t)(t - 1) * HID * HID, A, powers + (size_t)t * HID * HID);

  // K3: per-chunk local scans (parallel over 128 chunks)
  local_scan_kernel<<<NCHUNK, blk, 0, stream>>>(hs, A);

  // K4: sequential carry recurrence (128 steps, one workgroup)
  carry_kernel<<<1, blk, 0, stream>>>(hs, powers + (size_t)(L - 1) * HID * HID, carr);

  // K5: broadcast carries through the chunks (parallel)
  fixup_kernel<<<dim3(NCHUNK - 1, L), blk, 0, stream>>>(hs, powers, carr);

  // K6: ys = hs @ C^T
  gemm_wt_kernel<<<dim3(MROWS / 32, 2), blk, 0, stream>>>(hs, C, ys);
}